// arrayPC_62294205662027
// MI455X (gfx1250) — compile-verified
//
#include <hip/hip_runtime.h>
#include <hip/hip_bf16.h>
#include <math.h>

// Problem constants (from reference): N=512, K=513, B=65536
#define NN      512
#define KK      513
#define BB      65536
#define STEPS   511              // N-1 rows in log-table
#define ROWF    (KK * 2)         // 1026 floats per table row
#define LW_FLOATS (STEPS * KK * 2)   // 524286 floats (~2 MB) -> L2 resident
#define LOGE_OFF  524288             // float offset of logE inside workspace

#if __has_builtin(__builtin_amdgcn_global_load_async_to_lds_b128) && \
    __has_builtin(__builtin_amdgcn_s_wait_asynccnt)
#define USE_ASYNC_LDS 1
// Builtin signature (from clang diagnostic): (v4i __device__*, v4i AS3*, Ii, Ii)
typedef int v4i __attribute__((vector_size(16)));
typedef __attribute__((address_space(1))) v4i* gv4i_t;   // global int4*
typedef __attribute__((address_space(3))) v4i* lv4i_t;   // LDS int4*
#endif

__device__ __forceinline__ unsigned ballot32(bool p) {
#if __has_builtin(__builtin_amdgcn_ballot_w32)
    return __builtin_amdgcn_ballot_w32(p);
#else
    return (unsigned)__ballot(p);
#endif
}

// ---------------------------------------------------------------------------
// Main kernel: one wave32 per batch row.
//   - async-copy the whole 2KB x-row into LDS (4 x b128/lane, NT hint), one wait
//   - binary cumsum via ballot + chained mbcnt_lo: 4 VALU ops for a 128-wide
//     exclusive scan, chunk totals via scalar popcounts (base stays uniform)
//   - 4 gathers/lane from the L2-resident log-table, accumulate
//   - butterfly reduce, add logE[g_total] (scalar), lane 0 stores
// ---------------------------------------------------------------------------
__global__ __launch_bounds__(256) void arraypc_main(const int* __restrict__ x,
                                                    const float* __restrict__ Lw,
                                                    const float* __restrict__ logE,
                                                    float* __restrict__ out) {
    __shared__ int smem[8 * NN];                 // 8 waves * 512 ints = 16 KB
    const int lane = threadIdx.x & 31;
    const int wid  = threadIdx.x >> 5;
    const int b    = blockIdx.x * 8 + wid;       // one row per wave
    const int* xb  = x + (size_t)b * NN;
    int* sx        = smem + wid * NN;

#ifdef USE_ASYNC_LDS
    #pragma unroll
    for (int j = 0; j < 4; ++j) {
        __builtin_amdgcn_global_load_async_to_lds_b128(
            (gv4i_t)(xb + j * 128 + lane * 4),
            (lv4i_t)(sx + j * 128 + lane * 4),
            0, /*cpol: TH=NT (read-once stream)*/ 1);
    }
    __builtin_amdgcn_s_wait_asynccnt(0);
    __asm volatile("" ::: "memory");
#else
    #pragma unroll
    for (int j = 0; j < 4; ++j)
        reinterpret_cast<int4*>(sx)[j * 32 + lane] =
            reinterpret_cast<const int4*>(xb)[j * 32 + lane];
#endif

    float acc = 0.0f;
    int base = 0;                                // uniform cumsum carry
    #pragma unroll
    for (int t = 0; t < 4; ++t) {
        int4 v = reinterpret_cast<const int4*>(sx)[t * 32 + lane];
        // wave-wide masks of the binary inputs (one per int4 slot)
        unsigned m0 = ballot32(v.x != 0);
        unsigned m1 = ballot32(v.y != 0);
        unsigned m2 = ballot32(v.z != 0);
        unsigned m3 = ballot32(v.w != 0);
        // exclusive prefix over all 128 chunk elements in 4 mbcnt ops
        unsigned cb = __builtin_amdgcn_mbcnt_lo(m0, 0u);
        cb = __builtin_amdgcn_mbcnt_lo(m1, cb);
        cb = __builtin_amdgcn_mbcnt_lo(m2, cb);
        cb = __builtin_amdgcn_mbcnt_lo(m3, cb);
        int excl = base + (int)cb;
        int g0 = excl + v.x;     // inclusive g at element i0
        int g1 = g0 + v.y;
        int g2 = g1 + v.z;
        int g3 = g2 + v.w;
        int i0 = t * 128 + 4 * lane;   // global element index of v.x
        // element i contributes Lw[(i-1)*1026 + g_i*2 + x_i]  (skip i==0)
        if (i0 > 0) acc += Lw[(i0 - 1) * ROWF + 2 * g0 + v.x];
        acc += Lw[(i0 + 0) * ROWF + 2 * g1 + v.y];
        acc += Lw[(i0 + 1) * ROWF + 2 * g2 + v.z];
        acc += Lw[(i0 + 2) * ROWF + 2 * g3 + v.w];
        // chunk total: scalar popcounts of uniform masks
        base += __popc(m0) + __popc(m1) + __popc(m2) + __popc(m3);
    }
    #pragma unroll
    for (int off = 16; off; off >>= 1) acc += __shfl_xor(acc, off, 32);
    if (lane == 0) out[b] = acc + logE[base];
}

// ---------------------------------------------------------------------------
// Kernel 1: Lw[i, j, c] = log( W_full[i, j, c] )  (exact closed form of the
// softmax * ADJ1 + ADJ2 construction; -inf slots are never gathered).
// ---------------------------------------------------------------------------
__global__ __launch_bounds__(256) void prep_lw(const float* __restrict__ W,
                                               float* __restrict__ Lw) {
    int tid = blockIdx.x * blockDim.x + threadIdx.x;   // tid = i*513 + j
    if (tid >= STEPS * KK) return;
    int i = tid / KK;       // step 0..510
    int j = tid - i * KK;   // column 0..512
    float2 r;
    if (j == 0) {
        r.x = 0.0f;          // append column: log(1)
        r.y = -INFINITY;     // log(0), never gathered
    } else {
        int base = (i * (KK - 1) + (j - 1)) * 2;   // W is (511, 512, 2)
        float a = W[base + 0];
        float b = W[base + 1];
        float m = fmaxf(a, b);
        float lse = m + log1pf(expf(-fabsf(a - b)));   // log(e^a + e^b)
        int f0 = (j - 1) * 2;
        int th = 2 * i + 2;
        r.x = (f0     < th) ? (a - lse) : ((f0     == th + 1) ? 0.0f : -INFINITY);
        r.y = (f0 + 1 < th) ? (b - lse) : ((f0 + 1 == th + 1) ? 0.0f : -INFINITY);
    }
    reinterpret_cast<float2*>(Lw)[tid] = r;   // floats 2*tid = i*1026 + j*2
}

// ---------------------------------------------------------------------------
// Kernel 2: logE[k] = endW[k] - logsumexp(endW)   (K = 513, one wave32)
// ---------------------------------------------------------------------------
__global__ __launch_bounds__(32) void prep_end(const float* __restrict__ endW,
                                               float* __restrict__ logE) {
    int lane = threadIdx.x;
    float m = -INFINITY;
    for (int k = lane; k < KK; k += 32) m = fmaxf(m, endW[k]);
    for (int off = 16; off; off >>= 1) m = fmaxf(m, __shfl_xor(m, off, 32));
    float s = 0.0f;
    for (int k = lane; k < KK; k += 32) s += expf(endW[k] - m);
    for (int off = 16; off; off >>= 1) s += __shfl_xor(s, off, 32);
    float lse = m + logf(s);
    for (int k = lane; k < KK; k += 32) logE[k] = endW[k] - lse;
}

// ---------------------------------------------------------------------------
extern "C" void kernel_launch(void* const* d_in, const int* in_sizes, int n_in,
                              void* d_out, int out_size, void* d_ws, size_t ws_size,
                              hipStream_t stream) {
    const int*   x    = (const int*)d_in[0];    // (B, N) int32
    const float* W    = (const float*)d_in[1];  // (N-1, K-1, 2) f32
    const float* endW = (const float*)d_in[2];  // (1, K) f32
    float* out  = (float*)d_out;                // (B, 1) f32
    float* Lw   = (float*)d_ws;                 // 524286 floats
    float* logE = (float*)d_ws + LOGE_OFF;      // 513 floats

    // build log tables (tiny) then the bandwidth-bound main pass
    prep_lw<<<(STEPS * KK + 255) / 256, 256, 0, stream>>>(W, Lw);
    prep_end<<<1, 32, 0, stream>>>(endW, logE);
    arraypc_main<<<BB / 8, 256, 0, stream>>>(x, Lw, logE, out);
}